// SimpleCat_4398046511384
// MI455X (gfx1250) — compile-verified
//
#include <hip/hip_runtime.h>

// ---------------------------------------------------------------------------
// Embedding gather (B=4096, S=100, V=100000, D=300, MD=50) — pure bandwidth.
// sent_vec: 491 MB written via wave-per-row b128 async DMA (global -> LDS ->
// global, tracked with ASYNCcnt / s_wait_asynccnt). Outputs are streamed with
// non-temporal hints so the 120 MB word_embed table stays resident in the
// 192 MB L2 (the only data with reuse in this problem).
// ---------------------------------------------------------------------------

#define AS1 __attribute__((address_space(1)))
#define AS3 __attribute__((address_space(3)))

#ifndef __has_builtin
#define __has_builtin(x) 0
#endif

#if __has_builtin(__builtin_amdgcn_global_load_async_to_lds_b128) && \
    __has_builtin(__builtin_amdgcn_global_store_async_from_lds_b128) && \
    __has_builtin(__builtin_amdgcn_s_wait_asynccnt)
#define HAS_ASYNC_LDS 1
#else
#define HAS_ASYNC_LDS 0
#endif

// Native vector types (HIP float2/float4 are classes; builtins reject them).
typedef int   v4i __attribute__((vector_size(16)));
typedef float f4  __attribute__((vector_size(16)));
typedef float f2  __attribute__((vector_size(8)));

typedef AS1 v4i* g4ptr;   // global pointer to 16B chunk
typedef AS3 v4i* l4ptr;   // LDS pointer to 16B chunk

static constexpr int S_LEN = 100;   // sequence length
static constexpr int D_DIM = 300;   // word embed dim  (1200 B/row, 16B aligned)
static constexpr int M_DIM = 50;    // mask embed dim  (200 B/row, 8B aligned)
static constexpr int CH    = D_DIM / 4;  // 75 x 16B chunks per word row
static constexpr int WPB   = 8;          // waves per block (256 threads, wave32)

__device__ __forceinline__ g4ptr as_global4(const void* p) {
  return (g4ptr)(unsigned long long)p;
}
__device__ __forceinline__ l4ptr as_lds4(const void* p) {
  // generic LDS pointer: low 32 bits are the LDS byte offset (aperture form)
  return (l4ptr)(unsigned int)(unsigned long long)p;
}

// --------------------------- sent_vec gather -------------------------------
__global__ void __launch_bounds__(WPB * 32)
gather_word_rows(const int* __restrict__ sent,
                 const float* __restrict__ word_embed,
                 float* __restrict__ out, int n_rows) {
  const int lane = threadIdx.x & 31;
  const int wv   = threadIdx.x >> 5;
  const long row = (long)blockIdx.x * WPB + wv;
  if (row >= n_rows) return;

  const int idx = sent[row];
  const f4* __restrict__ src = (const f4*)(word_embed + (long)idx * D_DIM);
  f4* __restrict__ dst       = (f4*)(out + row * (long)D_DIM);

#if HAS_ASYNC_LDS
  // CDNA5 async DMA path: global -> LDS -> global, tracked by ASYNCcnt.
  __shared__ f4 stage[WPB][CH];
  for (int c = lane; c < CH; c += 32)
    __builtin_amdgcn_global_load_async_to_lds_b128(
        as_global4(src + c), as_lds4(&stage[wv][c]), 0, 0);
  __builtin_amdgcn_s_wait_asynccnt(0);
  for (int c = lane; c < CH; c += 32)
    __builtin_amdgcn_global_store_async_from_lds_b128(
        as_global4(dst + c), as_lds4(&stage[wv][c]), 0, 0);
  __builtin_amdgcn_s_wait_asynccnt(0);
#else
  // b128 register copy; NT store keeps L2 free for the embed table.
  for (int c = lane; c < CH; c += 32) {
    f4 v = src[c];
    __builtin_nontemporal_store(v, dst + c);
  }
#endif
}

// --------------------------- mask_vec gather -------------------------------
__global__ void __launch_bounds__(256)
gather_mask_rows(const int* __restrict__ mask,
                 const float* __restrict__ mask_embed,
                 float* __restrict__ out, long n2) {
  const long i2 = (long)blockIdx.x * blockDim.x + threadIdx.x;
  if (i2 >= n2) return;
  constexpr int HM = M_DIM / 2;          // 25 x 8B chunks per row
  const int cell = (int)(i2 / HM);
  const int j2   = (int)(i2 - (long)cell * HM);
  const int m    = mask[cell];           // 0 or 1; table lives in L0
  const f2* me2  = (const f2*)mask_embed;
  f2 v = me2[m * HM + j2];
  __builtin_nontemporal_store(v, ((f2*)out) + i2);
}

// --------------------------- position weight -------------------------------
__global__ void __launch_bounds__(256)
position_weight_kernel(const int* __restrict__ mask,
                       const int* __restrict__ power_p,
                       float* __restrict__ out, int B) {
  const int lane = threadIdx.x & 31;
  const int wv   = threadIdx.x >> 5;
  const int b    = blockIdx.x * (blockDim.x >> 5) + wv;
  if (b >= B) return;

  const int* __restrict__ mrow = mask + (long)b * S_LEN;
  int tmin = S_LEN;   // first index where mask==1 (argmax for 0/1 data)
  int Lsum = 0;       // run length
#pragma unroll
  for (int k = 0; k < 4; ++k) {
    const int p = lane + 32 * k;
    const int m = (p < S_LEN) ? mrow[p] : 0;
    Lsum += m;
    if (m && p < tmin) tmin = p;
  }
  // wave32 reductions
#pragma unroll
  for (int off = 16; off > 0; off >>= 1) {
    tmin = min(tmin, __shfl_xor(tmin, off, 32));
    Lsum += __shfl_xor(Lsum, off, 32);
  }
  const int t = tmin, L = Lsum;
  const float pw = (float)power_p[0];

#pragma unroll
  for (int k = 0; k < 4; ++k) {
    const int p = lane + 32 * k;
    if (p >= S_LEN) continue;
    float w;
    if (p >= t && p <= t + L) {
      w = 1.0f;
    } else if (p < t) {
      int d = t - p; d = d > 99 ? 99 : d;
      w = __powf((100.0f - (float)d) * 0.01f, pw);
    } else {
      int d = p - t - L; d = d < 0 ? 0 : (d > 99 ? 99 : d);
      w = __powf((100.0f - (float)d) * 0.01f, pw);
    }
    __builtin_nontemporal_store(w, out + (long)b * S_LEN + p);
  }
}

// ---------------------------------------------------------------------------
extern "C" void kernel_launch(void* const* d_in, const int* in_sizes, int n_in,
                              void* d_out, int out_size, void* d_ws,
                              size_t ws_size, hipStream_t stream) {
  const int*   sent       = (const int*)d_in[0];
  const int*   mask       = (const int*)d_in[1];
  const float* word_embed = (const float*)d_in[2];
  const float* mask_embed = (const float*)d_in[3];
  const int*   power      = (const int*)d_in[4];

  const int BS = in_sizes[0];      // B*S = 409600
  const int B  = BS / S_LEN;

  float* out_sent = (float*)d_out;                       // BS*D floats
  float* out_mask = out_sent + (long)BS * D_DIM;         // BS*MD floats
  float* out_w    = out_mask + (long)BS * M_DIM;         // BS floats

  {
    const int blocks = (BS + WPB - 1) / WPB;             // wave-per-row
    gather_word_rows<<<blocks, WPB * 32, 0, stream>>>(sent, word_embed,
                                                      out_sent, BS);
  }
  {
    const long n2 = (long)BS * (M_DIM / 2);
    const int blocks = (int)((n2 + 255) / 256);
    gather_mask_rows<<<blocks, 256, 0, stream>>>(mask, mask_embed, out_mask,
                                                 n2);
  }
  {
    const int blocks = (B + 7) / 8;
    position_weight_kernel<<<blocks, 256, 0, stream>>>(mask, power, out_w, B);
  }
}